// Scatter_5574867550244
// MI455X (gfx1250) — compile-verified
//
#include <hip/hip_runtime.h>
#include <hip/hip_bf16.h>
#include <math.h>

#define NNODES  20000
#define CIN     64
#define NEDGE   640000
#define NGRAPH  32
#define DCOLS   704          // 11 * 64
#define OUTCOLS 2816         // 4 * 704

typedef __attribute__((ext_vector_type(2))) float v2f;
typedef __attribute__((ext_vector_type(8))) float v8f;

// ---------------- utility kernels ----------------
__global__ void k_zero_f32(float* p, int n) {
  int i = blockIdx.x * blockDim.x + threadIdx.x;
  if (i < n) p[i] = 0.0f;
}
__global__ void k_zero_i32(int* p, int n) {
  int i = blockIdx.x * blockDim.x + threadIdx.x;
  if (i < n) p[i] = 0;
}
__global__ void k_copy_v4(float4* dst, const float4* src, int n4) {
  int i = blockIdx.x * blockDim.x + threadIdx.x;
  if (i < n4) dst[i] = src[i];
}
__global__ void k_half_copy_v4(float4* dst, const float4* src, int n4) {
  int i = blockIdx.x * blockDim.x + threadIdx.x;
  if (i < n4) {
    float4 v = src[i];
    v.x *= 0.5f; v.y *= 0.5f; v.z *= 0.5f; v.w *= 0.5f;
    dst[i] = v;
  }
}

// ---------------- degree / graph offsets ----------------
__global__ void k_deg_count(const int* col, float* deg, int E) {
  int e = blockIdx.x * blockDim.x + threadIdx.x;
  if (e < E) atomicAdd(&deg[col[e]], 1.0f);
}
__global__ void k_deg_inv(const float* deg, float* dinv, int n) {
  int i = blockIdx.x * blockDim.x + threadIdx.x;
  if (i < n) dinv[i] = (deg[i] > 0.0f) ? (1.0f / deg[i]) : 0.0f;
}
__global__ void k_batch_count(const int* batch, int* gcnt, int n) {
  int i = blockIdx.x * blockDim.x + threadIdx.x;
  if (i < n) atomicAdd(&gcnt[batch[i]], 1);
}
__global__ void k_scan_starts(const int* gcnt, int* starts) {
  if (threadIdx.x == 0 && blockIdx.x == 0) {
    int s = 0;
    starts[0] = 0;
    for (int g = 0; g < NGRAPH; ++g) { s += gcnt[g]; starts[g + 1] = s; }
  }
}

// ---------------- phase init: h0 = x, data[:,0:64] = x ----------------
__global__ void k_init_phase1(const float* x, float* hA, float* data, int n64) {
  int i = blockIdx.x * blockDim.x + threadIdx.x;
  if (i < n64) {
    int nn = i >> 6, c = i & 63;
    float v = x[i];
    hA[i] = v;
    data[(size_t)nn * DCOLS + c] = v;
  }
}

// ------ edge diffusion: dst[col] += 0.5 * dinv[row] * src[row], 4 ch/thread ------
template <int W>   // W = row width in floats (64 or 256); W/4 quads per edge
__global__ void k_edge_diffuse_v4(float* dst, const float* src, const int* row,
                                  const int* col, const float* dinv, int E) {
  constexpr int QSH = (W == 64) ? 4 : 6;     // log2(W/4)
  long long idx = (long long)blockIdx.x * blockDim.x + threadIdx.x;
  if (idx >= ((long long)E << QSH)) return;
  int e = (int)(idx >> QSH);
  int q = (int)(idx & ((W / 4) - 1));
  int r = row[e];
  int d = col[e];
  float s = 0.5f * dinv[r];
  float4 hv = *(const float4*)(src + (size_t)r * W + (size_t)q * 4);
  float* dp = dst + (size_t)d * W + (size_t)q * 4;
  atomicAdd(dp + 0, s * hv.x);
  atomicAdd(dp + 1, s * hv.y);
  atomicAdd(dp + 2, s * hv.z);
  atomicAdd(dp + 3, s * hv.w);
}

// ---------------- wavelet diffs ----------------
// phase1: s1[n,c,w] = |cur - snap|, also data[n,(1+w)*64+c]; snap <- cur
__global__ void k_diff1(const float* cur, float* snap, float* data, float* s1,
                        int w, int n64) {
  int i = blockIdx.x * blockDim.x + threadIdx.x;
  if (i < n64) {
    float cv = cur[i];
    float v = fabsf(cv - snap[i]);
    snap[i] = cv;
    int nn = i >> 6, c = i & 63;
    data[(size_t)nn * DCOLS + (size_t)(1 + w) * 64 + c] = v;
    s1[((size_t)nn << 8) + ((size_t)c << 2) + w] = v;  // layout (n, c*4 + w) = lv2 h0
  }
}
// phase2: t[n,w,c,a] = |cur - snap| over (N,256); write selected FENG channels
__global__ void k_diff2(const float* cur, float* snap, float* data, int w, int n256) {
  int i = blockIdx.x * blockDim.x + threadIdx.x;
  if (i < n256) {
    float cv = cur[i];
    float v = fabsf(cv - snap[i]);
    snap[i] = cv;
    int nn = i >> 8;
    int j = i & 255;
    int c = j >> 2;
    int a = j & 3;
    int k = -1;
    if (w == 1)      { if (a == 0) k = 5; }                         // FENG idx 4
    else if (w == 2) { if (a == 0) k = 6; else if (a == 1) k = 7; } // 8, 9
    else if (w == 3) { if (a < 3)  k = 8 + a; }                     // 12, 13, 14
    if (k >= 0) data[(size_t)nn * DCOLS + (size_t)k * 64 + c] = v;
  }
}

// ---------------- per-graph column sums via WMMA (all-ones A) ----------------
// One wave per (graph, 32-column tile): two accumulators, two WMMAs per 4 rows.
// D = ones(16x4) x B(4x16) + C sums 4 node-rows per instruction; every D row is
// the column sum, so the result is invariant to the A/K register layout. Main
// loop is bounds-check-free (wave-uniform trip count); single masked tail.
__device__ __forceinline__ v8f wmma_acc_f32(v2f a, v2f b, v8f c) {
#if defined(__gfx1250__)
  return __builtin_amdgcn_wmma_f32_16x16x4_f32(false, a, false, b,
                                               (short)0, c, false, false);
#else
  c[0] += b.x + b.y;   // host-pass placeholder (never executed on device)
  return c;
#endif
}

__global__ void k_graph_mean_wmma(const float* data, const int* starts, float* mean) {
  int tile = blockIdx.x;       // 0..21 (32 columns each)
  int g = blockIdx.y;          // 0..31
  int lane = threadIdx.x;      // wave32
  int start = starts[g], end = starts[g + 1];
  int col0 = tile * 32 + (lane & 15);
  int col1 = col0 + 16;
  int kb = (lane < 16) ? 0 : 2;

  v8f acc0 = {0.f, 0.f, 0.f, 0.f, 0.f, 0.f, 0.f, 0.f};
  v8f acc1 = acc0;
  v2f a;  a.x = 1.0f; a.y = 1.0f;

  int cnt = end - start;
  int nfull = start + (cnt & ~3);
  const float* p = data + (size_t)(start + kb) * DCOLS;   // rows kb, kb+1 of tile
  for (int n0 = start; n0 < nfull; n0 += 4) {
    v2f b0, b1;
    b0.x = p[col0];         b1.x = p[col1];
    b0.y = p[DCOLS + col0]; b1.y = p[DCOLS + col1];
    p += (size_t)4 * DCOLS;
    acc0 = wmma_acc_f32(a, b0, acc0);
    acc1 = wmma_acc_f32(a, b1, acc1);
  }
  if (nfull < end) {                                      // wave-uniform condition
    int r0 = nfull + kb, r1 = r0 + 1;
    v2f b0, b1;
    b0.x = (r0 < end) ? data[(size_t)r0 * DCOLS + col0] : 0.0f;
    b0.y = (r1 < end) ? data[(size_t)r1 * DCOLS + col0] : 0.0f;
    b1.x = (r0 < end) ? data[(size_t)r0 * DCOLS + col1] : 0.0f;
    b1.y = (r1 < end) ? data[(size_t)r1 * DCOLS + col1] : 0.0f;
    acc0 = wmma_acc_f32(a, b0, acc0);
    acc1 = wmma_acc_f32(a, b1, acc1);
  }
  float fc = (float)cnt;
  if (fc < 1.0f) fc = 1.0f;
  // C/D layout: VGPR0, lanes 0-15 -> M=0, N=lane
  if (lane < 16) {
    mean[(size_t)g * DCOLS + col0] = acc0[0] / fc;
    mean[(size_t)g * DCOLS + col1] = acc1[0] / fc;
  }
}

// ---------------- per-graph central moment sums (pass 2) ----------------
__global__ void k_graph_moments(const float* data, const float* mean,
                                const int* starts, float* msum) {
  __shared__ float sh[3][4][64];
  int g = blockIdx.y;
  int base = blockIdx.x * 64;   // 11 tiles of 64 cols
  int t = threadIdx.x;
  int c = t & 63;
  int stripe = t >> 6;          // 4 node stripes
  int start = starts[g], end = starts[g + 1];
  float m = mean[(size_t)g * DCOLS + base + c];
  float s2 = 0.f, s3 = 0.f, s4 = 0.f;
  for (int n = start + stripe; n < end; n += 4) {
    float d = data[(size_t)n * DCOLS + base + c] - m;
    float d2 = d * d;
    s2 += d2; s3 += d2 * d; s4 += d2 * d2;
  }
  sh[0][stripe][c] = s2;
  sh[1][stripe][c] = s3;
  sh[2][stripe][c] = s4;
  __syncthreads();
  if (stripe == 0) {
    s2 = sh[0][0][c] + sh[0][1][c] + sh[0][2][c] + sh[0][3][c];
    s3 = sh[1][0][c] + sh[1][1][c] + sh[1][2][c] + sh[1][3][c];
    s4 = sh[2][0][c] + sh[2][1][c] + sh[2][2][c] + sh[2][3][c];
    size_t o = (size_t)g * DCOLS + base + c;
    size_t P = (size_t)NGRAPH * DCOLS;
    msum[o] = s2; msum[P + o] = s3; msum[2 * P + o] = s4;
  }
}

// ---------------- finalize: mean | m2 | skew | kurt ----------------
__global__ void k_finalize(const float* mean, const float* msum,
                           const int* starts, float* out) {
  int i = blockIdx.x * blockDim.x + threadIdx.x;
  if (i < NGRAPH * DCOLS) {
    int g = i / DCOLS;
    int c = i - g * DCOLS;
    float cnt = (float)(starts[g + 1] - starts[g]);
    if (cnt < 1.0f) cnt = 1.0f;
    size_t P = (size_t)NGRAPH * DCOLS;
    float mn = mean[i];
    float m2 = msum[i] / cnt;
    float m3 = msum[P + i] / cnt;
    float m4 = msum[2 * P + i] / cnt;
    float sk = m3 / (m2 * sqrtf(m2));
    sk = (sk > 1e15f) ? 0.0f : sk;          // inf > 1e15 -> 0 ; NaN compares false
    sk = (sk != sk) ? 0.0f : sk;            // NaN -> 0
    float ku = m4 / (m2 * m2) - 3.0f;
    ku = (ku > 1e15f) ? -3.0f : ku;
    ku = (ku != ku) ? -3.0f : ku;
    size_t ob = (size_t)g * OUTCOLS;
    out[ob + c]        = mn;
    out[ob + 704 + c]  = m2;
    out[ob + 1408 + c] = sk;
    out[ob + 2112 + c] = ku;
  }
}

__global__ void k_copy_wavelet(const float* wav, float* out, int n) {
  int i = blockIdx.x * blockDim.x + threadIdx.x;
  if (i < n) out[i] = wav[i];
}

// ---------------- host orchestration ----------------
static inline int blks(long long n, int b) { return (int)((n + b - 1) / b); }

extern "C" void kernel_launch(void* const* d_in, const int* in_sizes, int n_in,
                              void* d_out, int out_size, void* d_ws, size_t ws_size,
                              hipStream_t stream) {
  const float* x     = (const float*)d_in[0];
  const int*   ei    = (const int*)d_in[1];
  const int*   batch = (const int*)d_in[2];
  const float* wav   = (const float*)d_in[3];
  const int N = in_sizes[0] / CIN;
  const int E = in_sizes[1] / 2;
  const int Wn = in_sizes[3];
  const int* row = ei;
  const int* col = ei + E;

  // workspace carve-up (256B aligned regions)
  char* w = (char*)d_ws;
  size_t off = 0;
  auto carve = [&](size_t bytes) { void* p = w + off; off = (off + bytes + 255) & ~(size_t)255; return p; };
  float* deg    = (float*)carve((size_t)N * 4);
  float* dinv   = (float*)carve((size_t)N * 4);
  int*   gcnt   = (int*)carve(NGRAPH * 4);
  int*   starts = (int*)carve((NGRAPH + 1) * 4);
  float* mean   = (float*)carve((size_t)NGRAPH * DCOLS * 4);
  float* msum   = (float*)carve((size_t)3 * NGRAPH * DCOLS * 4);
  float* data   = (float*)carve((size_t)N * DCOLS * 4);
  float* s1     = (float*)carve((size_t)N * 256 * 4);
  float* hA     = (float*)carve((size_t)N * 256 * 4);
  float* hB     = (float*)carve((size_t)N * 256 * 4);
  float* snap   = (float*)carve((size_t)N * 256 * 4);
  (void)ws_size; (void)n_in;

  const int B = 256;
  const int n64 = N * 64, n256 = N * 256;

  // degrees and graph offsets
  k_zero_f32<<<blks(N, B), B, 0, stream>>>(deg, N);
  k_deg_count<<<blks(E, B), B, 0, stream>>>(col, deg, E);
  k_deg_inv<<<blks(N, B), B, 0, stream>>>(deg, dinv, N);
  k_zero_i32<<<1, 64, 0, stream>>>(gcnt, NGRAPH);
  k_batch_count<<<blks(N, B), B, 0, stream>>>(batch, gcnt, N);
  k_scan_starts<<<1, 32, 0, stream>>>(gcnt, starts);

  // ---- phase 1: lv1 diffusion, width 64, h0 = x ----
  k_init_phase1<<<blks(n64, B), B, 0, stream>>>(x, hA, data, n64);
  {
    const float* src = hA;
    float* bufs[2] = {hB, hA};
    for (int step = 1; step <= 16; ++step) {
      float* dst = bufs[(step - 1) & 1];
      k_half_copy_v4<<<blks(n64 / 4, B), B, 0, stream>>>((float4*)dst, (const float4*)src, n64 / 4);
      k_edge_diffuse_v4<64><<<blks((long long)E * 16, B), B, 0, stream>>>(dst, src, row, col, dinv, E);
      src = dst;
      if (step == 1)       k_copy_v4<<<blks(n64 / 4, B), B, 0, stream>>>((float4*)snap, (const float4*)src, n64 / 4);
      else if (step == 2)  k_diff1<<<blks(n64, B), B, 0, stream>>>(src, snap, data, s1, 0, n64);
      else if (step == 4)  k_diff1<<<blks(n64, B), B, 0, stream>>>(src, snap, data, s1, 1, n64);
      else if (step == 8)  k_diff1<<<blks(n64, B), B, 0, stream>>>(src, snap, data, s1, 2, n64);
      else if (step == 16) k_diff1<<<blks(n64, B), B, 0, stream>>>(src, snap, data, s1, 3, n64);
    }
  }

  // ---- phase 2: lv2 diffusion, width 256, h0 = s1 (layout n, c*4+a) ----
  {
    const float* src = s1;
    float* bufs[2] = {hB, hA};
    for (int step = 1; step <= 16; ++step) {
      float* dst = bufs[(step - 1) & 1];
      k_half_copy_v4<<<blks(n256 / 4, B), B, 0, stream>>>((float4*)dst, (const float4*)src, n256 / 4);
      k_edge_diffuse_v4<256><<<blks((long long)E * 64, B), B, 0, stream>>>(dst, src, row, col, dinv, E);
      src = dst;
      if (step == 1)       k_copy_v4<<<blks(n256 / 4, B), B, 0, stream>>>((float4*)snap, (const float4*)src, n256 / 4);
      else if (step == 2)  k_diff2<<<blks(n256, B), B, 0, stream>>>(src, snap, data, 0, n256);
      else if (step == 4)  k_diff2<<<blks(n256, B), B, 0, stream>>>(src, snap, data, 1, n256);
      else if (step == 8)  k_diff2<<<blks(n256, B), B, 0, stream>>>(src, snap, data, 2, n256);
      else if (step == 16) k_diff2<<<blks(n256, B), B, 0, stream>>>(src, snap, data, 3, n256);
    }
  }

  // ---- per-graph statistics ----
  k_graph_mean_wmma<<<dim3(DCOLS / 32, NGRAPH), 32, 0, stream>>>(data, starts, mean);
  k_graph_moments<<<dim3(DCOLS / 64, NGRAPH), 256, 0, stream>>>(data, mean, starts, msum);
  k_finalize<<<blks(NGRAPH * DCOLS, B), B, 0, stream>>>(mean, msum, starts, (float*)d_out);
  k_copy_wavelet<<<1, 128, 0, stream>>>(wav, (float*)d_out + (out_size - Wn), Wn);
}